// CustomVGAEEncoder_65996467470909
// MI455X (gfx1250) — compile-verified
//
#include <hip/hip_runtime.h>
#include <hip/hip_bf16.h>

// CDNA5 / gfx1250 VGAE-GCN encoder.
//  - Dense transforms via V_WMMA_F32_16X16X4_F32 (exact f32, matches reference).
//  - Message passing: wave-per-edge gather + global_atomic_add_f32 scatter
//    (h fits in 192MB L2, so random traffic stays on-chip).
// Sizes: N=100000 (div by 32), IN=H=128, OUT=64, E=1.6M.

typedef __attribute__((ext_vector_type(2))) float v2f;
typedef __attribute__((ext_vector_type(8))) float v8f;

// ---------------------------------------------------------------- utilities
__global__ void k_fill(float* __restrict__ p, float v, long n) {
  long i = (long)blockIdx.x * blockDim.x + threadIdx.x;
  if (i < n) p[i] = v;
}

__global__ void k_deg(const int* __restrict__ dst, float* __restrict__ deg, int E) {
  int e = blockIdx.x * blockDim.x + threadIdx.x;
  if (e < E) atomicAdd(&deg[dst[e]], 1.0f);
}

__global__ void k_rsqrt(float* __restrict__ d, int n) {
  int i = blockIdx.x * blockDim.x + threadIdx.x;
  if (i < n) d[i] = rsqrtf(d[i]);
}

// elu in place (jax.nn.elu: x>0 ? x : expm1(x))
__global__ void k_elu(float* __restrict__ p, long n) {
  long i = (long)blockIdx.x * blockDim.x + threadIdx.x;
  if (i < n) { float v = p[i]; p[i] = v > 0.0f ? v : expm1f(v); }
}

// xc = elu(t2) + w_residual * x1   (in place into t2)
__global__ void k_resid(float* __restrict__ t2, const float* __restrict__ x1,
                        const float* __restrict__ wres, long n) {
  long i = (long)blockIdx.x * blockDim.x + threadIdx.x;
  if (i < n) {
    float v = t2[i];
    float e = v > 0.0f ? v : expm1f(v);
    t2[i] = e + wres[0] * x1[i];
  }
}

// ---------------------------------------------------------------- WMMA GEMM
// H[N x F] = X[N x 128] @ W[128 x F],  F in {128, 64}, K = 128.
// Block: 256 threads (8 wave32), 32 rows of X per block.
// LDS: W (128*F floats) + X tile (32*128 floats), dynamic shared.
// Each wave computes one/two 16x16 f32 tiles with 32 chained
// v_wmma_f32_16x16x4_f32 ops. Fragment layouts per ISA 7.12.2:
//   A (16x4):  lane m (0..15)/m+16; vgpr v holds A[m][4*kk + v + 2*half]
//   B (4x16):  lane n/half; vgpr v holds B[4*kk + v + 2*half][n]
//   C (16x16): vgpr j holds D[j + 8*half][n]
template <int F>
__global__ __launch_bounds__(256) void k_gemm(const float* __restrict__ X,
                                              const float* __restrict__ W,
                                              float* __restrict__ H) {
  extern __shared__ float smem[];
  float* sW = smem;              // 128 x F
  float* sX = smem + 128 * F;    // 32 x 128
  const int tid = threadIdx.x;
  const long r0 = (long)blockIdx.x * 32;

  // stage W (coalesced float4)
  for (int idx = tid; idx < 128 * F / 4; idx += 256)
    ((float4*)sW)[idx] = ((const float4*)W)[idx];
  // stage 32 contiguous rows of X
  const float4* Xv = (const float4*)(X + r0 * 128);
  for (int idx = tid; idx < 32 * 128 / 4; idx += 256)
    ((float4*)sX)[idx] = Xv[idx];
  __syncthreads();

  const int wid  = tid >> 5;
  const int lane = tid & 31;
  const int half = lane >> 4;
  const int mn   = lane & 15;
  const int ctiles = F / 16;

  for (int t = wid; t < 2 * ctiles; t += 8) {   // uniform per wave
    const int s = t / ctiles;                   // row strip (0/1)
    const int c = (t % ctiles) * 16;            // col tile base
    v8f acc = {};
    const float* xrow = sX + (16 * s + mn) * 128;
    #pragma unroll 4
    for (int kk = 0; kk < 128; kk += 4) {
      v2f a, b;
      a[0] = xrow[kk + 2 * half + 0];
      a[1] = xrow[kk + 2 * half + 1];
      b[0] = sW[(kk + 2 * half + 0) * F + c + mn];
      b[1] = sW[(kk + 2 * half + 1) * F + c + mn];
      acc = __builtin_amdgcn_wmma_f32_16x16x4_f32(
          /*neg_a=*/false, a, /*neg_b=*/false, b,
          /*c_mod=*/(short)0, acc, /*reuse_a=*/false, /*reuse_b=*/false);
    }
    float* out = H + (r0 + 16 * s) * F + c + mn;
    #pragma unroll
    for (int j = 0; j < 8; ++j)
      out[(long)(j + 8 * half) * F] = acc[j];
  }
}

// ------------------------------------------------------- propagation kernels
// out[i,f] = bias[f] + dinv[i]^2 * H[i,f]      (self-loop + bias, overwrites)
template <int F>
__global__ void k_prop_init(const float* __restrict__ H, const float* __restrict__ dinv,
                            const float* __restrict__ bias, float* __restrict__ out,
                            int n) {
  long i = (long)blockIdx.x * blockDim.x + threadIdx.x;  // over n * F/4
  long total = (long)n * (F / 4);
  if (i >= total) return;
  long node = i / (F / 4);
  int  fq   = (int)(i % (F / 4));
  float di = dinv[node];
  float sc = di * di;
  float4 h = ((const float4*)H)[i];
  float4 b = ((const float4*)bias)[fq];
  float4 o;
  o.x = b.x + sc * h.x;  o.y = b.y + sc * h.y;
  o.z = b.z + sc * h.z;  o.w = b.w + sc * h.w;
  ((float4*)out)[i] = o;
}

// out[dst] += H[src] * dinv[src]*dinv[dst]   — one wave32 per edge,
// lane covers F/32 features (float4 for F=128, float2 for F=64).
template <int F>
__global__ __launch_bounds__(256) void k_prop_edges(const float* __restrict__ H,
                                                    const int* __restrict__ ei,
                                                    const float* __restrict__ dinv,
                                                    float* __restrict__ out, int E) {
  const int wid  = threadIdx.x >> 5;
  const int lane = threadIdx.x & 31;
  long e = (long)blockIdx.x * 8 + wid;
  if (e >= E) return;
  int s = ei[e];
  int d = ei[(long)E + e];
  float nrm = dinv[s] * dinv[d];
  if (F == 128) {
    float4 h = ((const float4*)(H + (long)s * F))[lane];
    float* o = out + (long)d * F + lane * 4;
    atomicAdd(o + 0, h.x * nrm);
    atomicAdd(o + 1, h.y * nrm);
    atomicAdd(o + 2, h.z * nrm);
    atomicAdd(o + 3, h.w * nrm);
  } else {
    float2 h = ((const float2*)(H + (long)s * F))[lane];
    float* o = out + (long)d * F + lane * 2;
    atomicAdd(o + 0, h.x * nrm);
    atomicAdd(o + 1, h.y * nrm);
  }
}

// ------------------------------------------------------------------- driver
extern "C" void kernel_launch(void* const* d_in, const int* in_sizes, int n_in,
                              void* d_out, int out_size, void* d_ws, size_t ws_size,
                              hipStream_t stream) {
  const float* x    = (const float*)d_in[0];
  const int*   ei   = (const int*)  d_in[1];
  const float* W1   = (const float*)d_in[2];
  const float* b1   = (const float*)d_in[3];
  const float* W2   = (const float*)d_in[4];
  const float* b2   = (const float*)d_in[5];
  const float* Wmu  = (const float*)d_in[6];
  const float* bmu  = (const float*)d_in[7];
  const float* Wls  = (const float*)d_in[8];
  const float* bls  = (const float*)d_in[9];
  const float* wres = (const float*)d_in[10];

  const int n = in_sizes[0] / 128;   // 100000 (divisible by 32)
  const int E = in_sizes[1] / 2;     // 1600000

  // workspace layout (floats)
  float* dinv = (float*)d_ws;                 // [n]    deg -> rsqrt(deg)
  float* h    = dinv + n;                     // [n*128] GEMM output (reused)
  float* x1   = h + (long)n * 128;            // [n*128] elu(layer1)
  float* t2   = x1 + (long)n * 128;           // [n*128] layer2 accum -> xc
  float* out_mu = (float*)d_out;              // [n*64]
  float* out_ls = out_mu + (long)n * 64;      // [n*64]

  const int gblocks = n / 32;
  const size_t shm128 = (size_t)(128 * 128 + 32 * 128) * sizeof(float); // 80 KB
  const size_t shm64  = (size_t)(128 * 64  + 32 * 128) * sizeof(float); // 48 KB
  const int eb = (E + 7) / 8;                 // wave-per-edge blocks
  const long nf128 = (long)n * 128;
  const long nq128 = (long)n * 32;            // n * 128/4
  const long nq64  = (long)n * 16;            // n * 64/4

  // ---- GCN normalization: deg = 1 + indeg(dst); dinv = rsqrt(deg)
  k_fill<<<(n + 255) / 256, 256, 0, stream>>>(dinv, 1.0f, n);
  k_deg<<<(E + 255) / 256, 256, 0, stream>>>(ei + E, dinv, E);
  k_rsqrt<<<(n + 255) / 256, 256, 0, stream>>>(dinv, n);

  // ---- layer 1: x1 = elu(prop(x @ W1) + b1)
  k_gemm<128><<<gblocks, 256, shm128, stream>>>(x, W1, h);
  k_prop_init<128><<<(nq128 + 255) / 256, 256, 0, stream>>>(h, dinv, b1, x1, n);
  k_prop_edges<128><<<eb, 256, 0, stream>>>(h, ei, dinv, x1, E);
  k_elu<<<(nf128 + 255) / 256, 256, 0, stream>>>(x1, nf128);

  // ---- layer 2: t2 = prop(x1 @ W2) + b2;  xc = elu(t2) + wres*x1 (into t2)
  k_gemm<128><<<gblocks, 256, shm128, stream>>>(x1, W2, h);
  k_prop_init<128><<<(nq128 + 255) / 256, 256, 0, stream>>>(h, dinv, b2, t2, n);
  k_prop_edges<128><<<eb, 256, 0, stream>>>(h, ei, dinv, t2, E);
  k_resid<<<(nf128 + 255) / 256, 256, 0, stream>>>(t2, x1, wres, nf128);

  // ---- mu head
  k_gemm<64><<<gblocks, 256, shm64, stream>>>(t2, Wmu, h);
  k_prop_init<64><<<(nq64 + 255) / 256, 256, 0, stream>>>(h, dinv, bmu, out_mu, n);
  k_prop_edges<64><<<eb, 256, 0, stream>>>(h, ei, dinv, out_mu, E);

  // ---- logstd head
  k_gemm<64><<<gblocks, 256, shm64, stream>>>(t2, Wls, h);
  k_prop_init<64><<<(nq64 + 255) / 256, 256, 0, stream>>>(h, dinv, bls, out_ls, n);
  k_prop_edges<64><<<eb, 256, 0, stream>>>(h, ei, dinv, out_ls, E);
}